// RNNT_17918603559345
// MI455X (gfx1250) — compile-verified
//
#include <hip/hip_runtime.h>
#include <math.h>

typedef float v2f __attribute__((ext_vector_type(2)));
typedef float v4f __attribute__((ext_vector_type(4)));
typedef float v8f __attribute__((ext_vector_type(8)));

#define H    512
#define FOURH 2048
#define BDIM 4
#define TT   256
#define UU   64
#define VV   1024
#define DIN  80
#define NWG  16   // workgroups in persistent LSTM recurrence

// ---------------------------------------------------------------- masks
__global__ __launch_bounds__(256) void mask_kernel(const int* __restrict__ lens,
                                                   float* __restrict__ m, int S) {
  int i = blockIdx.x * 256 + threadIdx.x;
  if (i < BDIM * S) m[i] = ((i % S) < lens[i / S]) ? 1.f : 0.f;
}

// ---------------------------------------------------------------- embedding gather
__global__ __launch_bounds__(256) void embed_kernel(const int* __restrict__ ids,
                                                    const float* __restrict__ emb,
                                                    float* __restrict__ out) {
  int r = blockIdx.x;                       // B*U rows
  int id = ids[r];
  const v2f* src = (const v2f*)(emb + (size_t)id * H);
  v2f* dst = (v2f*)(out + (size_t)r * H);
  dst[threadIdx.x] = src[threadIdx.x];      // 256 threads x float2 = 512 floats
}

// ---------------------------------------------------------------- fp32 WMMA GEMM
// C[m,n] = rowscale[m] * sum_k A[m,k]*W[n,k]  + b1[n] + b2[n]
// One wave per 16x16 tile; V_WMMA_F32_16X16X4_F32, K multiple of 4.
__global__ __launch_bounds__(32) void wmma_gemm_f32(
    const float* __restrict__ A, int lda,
    const float* __restrict__ W, int ldw,
    const float* __restrict__ b1, const float* __restrict__ b2,
    const float* __restrict__ rowscale,
    float* __restrict__ C, int ldc, int K) {
  int lane = threadIdx.x;
  int half = lane >> 4, r = lane & 15;
  const float* arow = A + (size_t)(blockIdx.y * 16 + r) * lda + 2 * half;
  const float* wrow = W + (size_t)(blockIdx.x * 16 + r) * ldw + 2 * half;
  v8f acc = {0.f, 0.f, 0.f, 0.f, 0.f, 0.f, 0.f, 0.f};
  int niter = K >> 2;
#pragma unroll 4
  for (int j = 0; j < niter; ++j) {
    float2 af = *(const float2*)(arow + 4 * j);
    float2 wf = *(const float2*)(wrow + 4 * j);
    v2f a = {af.x, af.y};
    v2f b = {wf.x, wf.y};
    acc = __builtin_amdgcn_wmma_f32_16x16x4_f32(false, a, false, b, (short)0,
                                                acc, false, false);
  }
  int n = blockIdx.x * 16 + r;
  float bias = (b1 ? b1[n] : 0.f) + (b2 ? b2[n] : 0.f);
#pragma unroll
  for (int i = 0; i < 8; ++i) {
    int m = blockIdx.y * 16 + i + 8 * half;
    float v = acc[i];
    if (rowscale) v *= rowscale[m];         // mask applied BEFORE bias (matches ref)
    C[(size_t)m * ldc + n] = v + bias;
  }
}

// ---------------------------------------------------------------- persistent LSTM recurrence
// gates[b,n] = Xproj[b,t,n] + sum_k h[b,k]*Whh[n,k]; PyTorch gate order (i,f,g,o).
// 16 WGs x 256 threads (8 waves). WG `wg` owns h columns [wg*32, wg*32+32).
// Wave w computes gate (w>>1), 16-col subtile (w&1). h lives zero-padded (M=16)
// in LDS; Whh streamed from L2 (4.2 MB, resident). Grid barrier per timestep.
__global__ __launch_bounds__(256) void lstm_persistent(
    const float* __restrict__ Xproj, const float* __restrict__ Whh,
    float* __restrict__ Y, float* __restrict__ hbuf,
    int* __restrict__ counters, int S) {
  __shared__ float h_lds[16][H + 4];        // bank-padded, rows 4..15 stay zero
  __shared__ float g_lds[4][16][32];        // gate tiles (only m<4 consumed)
  __shared__ float c_lds[BDIM][32];
  int tid = threadIdx.x;
  int wg = blockIdx.x;
  for (int i = tid; i < 16 * (H + 4); i += 256) ((float*)h_lds)[i] = 0.f;
  if (tid < BDIM * 32) c_lds[tid >> 5][tid & 31] = 0.f;
  __syncthreads();

  int wave = tid >> 5, lane = tid & 31;
  int half = lane >> 4, r = lane & 15;
  int gate = wave >> 1;
  int cbase = (wave & 1) * 16;
  int ncol = wg * 32 + cbase + r;           // column within H for this B-row
  const float* wrow = Whh + (size_t)(gate * H + ncol) * H + 2 * half;

  for (int t = 0; t < S; ++t) {
    v8f acc = {0.f, 0.f, 0.f, 0.f, 0.f, 0.f, 0.f, 0.f};
#pragma unroll 4
    for (int j = 0; j < H / 4; ++j) {
      int k = 4 * j + 2 * half;
      v2f a = {h_lds[r][k], h_lds[r][k + 1]};
      float2 wf = *(const float2*)(wrow + 4 * j);
      v2f b = {wf.x, wf.y};
      acc = __builtin_amdgcn_wmma_f32_16x16x4_f32(false, a, false, b, (short)0,
                                                  acc, false, false);
    }
#pragma unroll
    for (int i = 0; i < 8; ++i) g_lds[gate][i + 8 * half][cbase + r] = acc[i];
    __syncthreads();

    if (tid < BDIM * 32) {                  // LSTM cell, one thread per (b, col)
      int b = tid >> 5, col = tid & 31;
      int hcol = wg * 32 + col;
      const float* xp = Xproj + (size_t)(b * S + t) * FOURH;
      float gi = g_lds[0][b][col] + xp[0 * H + hcol];
      float gf = g_lds[1][b][col] + xp[1 * H + hcol];
      float gg = g_lds[2][b][col] + xp[2 * H + hcol];
      float go = g_lds[3][b][col] + xp[3 * H + hcol];
      float i_ = 1.f / (1.f + __expf(-gi));
      float f_ = 1.f / (1.f + __expf(-gf));
      float g_ = tanhf(gg);
      float o_ = 1.f / (1.f + __expf(-go));
      float c = f_ * c_lds[b][col] + i_ * g_;
      c_lds[b][col] = c;
      float h = o_ * tanhf(c);
      Y[(size_t)(b * S + t) * H + hcol] = h;
      hbuf[b * H + hcol] = h;
    }
    __threadfence();
    __syncthreads();
    if (tid == 0) {                         // device-scope split barrier on L2 atomic
      __hip_atomic_fetch_add(&counters[t], 1, __ATOMIC_RELEASE,
                             __HIP_MEMORY_SCOPE_AGENT);
      while (__hip_atomic_load(&counters[t], __ATOMIC_ACQUIRE,
                               __HIP_MEMORY_SCOPE_AGENT) < NWG)
        __builtin_amdgcn_s_sleep(1);
    }
    __syncthreads();
    __threadfence();
    for (int i = tid; i < BDIM * H; i += 256)   // all-gather new h into LDS
      h_lds[i >> 9][i & (H - 1)] = hbuf[i];
    __syncthreads();
  }
}

// ---------------------------------------------------------------- joint broadcast-add
// out[b,t,u,v] = enc_p[b,t,v] + dec_p[b,u,v] + cls_b[v]; 256 MiB streaming store.
__global__ __launch_bounds__(256) void joint_kernel(
    const float* __restrict__ enc_p, const float* __restrict__ dec_p,
    const float* __restrict__ cls_b, float* __restrict__ out) {
  int idx = blockIdx.x;                     // flat (b,t,u)
  int u = idx % UU;
  int bt = idx / UU;
  int b = bt / TT;
  int tid = threadIdx.x;
  const v4f* e = (const v4f*)(enc_p + (size_t)bt * VV);
  const v4f* d = (const v4f*)(dec_p + (size_t)(b * UU + u) * VV);
  const v4f* cb = (const v4f*)cls_b;
  v4f o = e[tid] + d[tid] + cb[tid];
  __builtin_nontemporal_store(o, (v4f*)out + (size_t)idx * (VV / 4) + tid);
}

// ---------------------------------------------------------------- launch
extern "C" void kernel_launch(void* const* d_in, const int* in_sizes, int n_in,
                              void* d_out, int out_size, void* d_ws, size_t ws_size,
                              hipStream_t stream) {
  (void)in_sizes; (void)n_in; (void)out_size; (void)ws_size;
  const float* enc_x    = (const float*)d_in[0];
  const int*   enc_lens = (const int*)d_in[1];
  const int*   dec_x    = (const int*)d_in[2];
  const int*   dec_lens = (const int*)d_in[3];
  // enc_lstm layers: d_in[4+4l + {Wih,Whh,bih,bhh}]
  const float* enc_lin_W = (const float*)d_in[20];
  const float* enc_lin_b = (const float*)d_in[21];
  const float* embed     = (const float*)d_in[22];
  // dec_lstm layers: d_in[23+4l + {Wih,Whh,bih,bhh}]
  const float* cls_W = (const float*)d_in[31];
  const float* cls_b = (const float*)d_in[32];
  float* out = (float*)d_out;

  // workspace carve-up (~19.6 MiB)
  float* ws = (float*)d_ws;
  size_t off = 0;
  float* XPROJ = ws + off; off += (size_t)BDIM * TT * FOURH;  // 2,097,152
  float* YA    = ws + off; off += (size_t)BDIM * TT * H;
  float* YB    = ws + off; off += (size_t)BDIM * TT * H;
  float* ENC_O = ws + off; off += (size_t)BDIM * TT * H;
  float* ENC_P = ws + off; off += (size_t)BDIM * TT * VV;
  float* DECIN = ws + off; off += (size_t)BDIM * UU * H;
  float* DEC_P = ws + off; off += (size_t)BDIM * UU * VV;
  float* HBUF  = ws + off; off += 4096;
  float* TMASK = ws + off; off += 1024;
  float* UMASK = ws + off; off += 1024;
  int*   CTRS  = (int*)(ws + off);                            // 2048 ints

  hipMemsetAsync(CTRS, 0, 2048 * sizeof(int), stream);        // capture-legal
  mask_kernel<<<dim3((BDIM * TT + 255) / 256), 256, 0, stream>>>(enc_lens, TMASK, TT);
  mask_kernel<<<dim3((BDIM * UU + 255) / 256), 256, 0, stream>>>(dec_lens, UMASK, UU);
  embed_kernel<<<dim3(BDIM * UU), 256, 0, stream>>>(dec_x, embed, DECIN);

  int ctr = 0;
  float* Ybuf[2] = {YA, YB};

  // ---- encoder: 4 LSTM layers ----
  const float* X = enc_x;
  for (int l = 0; l < 4; ++l) {
    const float* Wih = (const float*)d_in[4 + 4 * l + 0];
    const float* Whh = (const float*)d_in[4 + 4 * l + 1];
    const float* bih = (const float*)d_in[4 + 4 * l + 2];
    const float* bhh = (const float*)d_in[4 + 4 * l + 3];
    int d = l ? H : DIN;
    wmma_gemm_f32<<<dim3(FOURH / 16, (BDIM * TT) / 16), 32, 0, stream>>>(
        X, d, Wih, d, bih, bhh, nullptr, XPROJ, FOURH, d);
    lstm_persistent<<<dim3(NWG), 256, 0, stream>>>(XPROJ, Whh, Ybuf[l & 1],
                                                   HBUF, CTRS + ctr, TT);
    ctr += TT;
    X = Ybuf[l & 1];
  }
  // enc_o = (enc_h * tmask) @ enc_lin_W^T + b
  wmma_gemm_f32<<<dim3(H / 16, (BDIM * TT) / 16), 32, 0, stream>>>(
      X, H, enc_lin_W, H, enc_lin_b, nullptr, TMASK, ENC_O, H, H);
  // enc_p = enc_o @ cls_W[:, :H]^T
  wmma_gemm_f32<<<dim3(VV / 16, (BDIM * TT) / 16), 32, 0, stream>>>(
      ENC_O, H, cls_W, 2 * H, nullptr, nullptr, nullptr, ENC_P, VV, H);

  // ---- decoder: 2 LSTM layers ----
  X = DECIN;
  for (int l = 0; l < 2; ++l) {
    const float* Wih = (const float*)d_in[23 + 4 * l + 0];
    const float* Whh = (const float*)d_in[23 + 4 * l + 1];
    const float* bih = (const float*)d_in[23 + 4 * l + 2];
    const float* bhh = (const float*)d_in[23 + 4 * l + 3];
    wmma_gemm_f32<<<dim3(FOURH / 16, (BDIM * UU) / 16), 32, 0, stream>>>(
        X, H, Wih, H, bih, bhh, nullptr, XPROJ, FOURH, H);
    lstm_persistent<<<dim3(NWG), 256, 0, stream>>>(XPROJ, Whh, Ybuf[l & 1],
                                                   HBUF, CTRS + ctr, UU);
    ctr += UU;
    X = Ybuf[l & 1];
  }
  // dec_p = (dec_o * umask) @ cls_W[:, H:]^T
  wmma_gemm_f32<<<dim3(VV / 16, (BDIM * UU) / 16), 32, 0, stream>>>(
      X, H, cls_W + H, 2 * H, nullptr, nullptr, UMASK, DEC_P, VV, H);

  // ---- joint: 256 MiB streaming output ----
  joint_kernel<<<dim3(BDIM * TT * UU), 256, 0, stream>>>(ENC_P, DEC_P, cls_b, out);
}